// PCA_Block_3831110828227
// MI455X (gfx1250) — compile-verified
//
#include <hip/hip_runtime.h>
#include <hip/hip_bf16.h>

typedef __attribute__((ext_vector_type(2))) float v2f;
typedef __attribute__((ext_vector_type(8))) float v8f;

#define NB   32          // batches
#define NC   256         // channels (features)
#define HW   4096        // 64*64
#define NTOT (NB * HW)   // 131072 samples
#define KOUT 128         // top-k eigenvectors

// ---------------------------------------------------------------------------
// 1. Per-channel mean over all N = B*H*W samples.
// ---------------------------------------------------------------------------
__global__ __launch_bounds__(256) void mean_kernel(const float* __restrict__ data,
                                                   float* __restrict__ mu) {
  const int c = blockIdx.x;
  const int tid = threadIdx.x;
  float s = 0.f;
  for (int b = 0; b < NB; ++b) {
    const float* p = data + ((size_t)b * NC + c) * HW;
    for (int i = tid; i < HW; i += 256) s += p[i];
  }
  __shared__ float red[256];
  red[tid] = s;
  __syncthreads();
  for (int off = 128; off > 0; off >>= 1) {
    if (tid < off) red[tid] += red[tid + off];
    __syncthreads();
  }
  if (tid == 0) mu[c] = red[0] / (float)NTOT;
}

// ---------------------------------------------------------------------------
// 2a. Zero the Gram accumulator.
// ---------------------------------------------------------------------------
__global__ void zero_kernel(float* __restrict__ G) {
  G[(size_t)blockIdx.x * 256 + threadIdx.x] = 0.f;
}

// ---------------------------------------------------------------------------
// 2b. Raw Gram G = sum_n x[n,:] x[n,:]^T using V_WMMA_F32_16X16X4_F32.
//     Per batch b, X_b is C x HW channel-major, so G += X_b * X_b^T.
//     Block = 8 waves; block (bi,b): output rows [32*bi, 32*bi+32) x all 256
//     cols for batch b. LDS slab: 256 rows x 32 k-columns, padded stride 36.
// ---------------------------------------------------------------------------
#define KT 32
#define SLAB_STRIDE 36   // floats; 144B, 16B-aligned, conflict-free (gcd trick)

__global__ __launch_bounds__(256) void gram_kernel(const float* __restrict__ data,
                                                   float* __restrict__ G) {
  __shared__ float slab[256 * SLAB_STRIDE];

  const int bi   = blockIdx.x;       // 0..7  (pair of 16-row tiles)
  const int b    = blockIdx.y;       // 0..31 (batch)
  const int tid  = threadIdx.x;
  const int wave = tid >> 5;
  const int lane = tid & 31;
  const int laneM = lane & 15;
  const int koff  = (lane >> 4) * 2; // lanes 0-15 -> K 0,1 ; lanes 16-31 -> K 2,3

  const int iTile = wave >> 2;            // 0..1
  const int jBase = (wave & 3) * 4;       // tile-col group: 0,4,8,12
  const int rowA  = bi * 32 + iTile * 16 + laneM;

  v8f acc[4] = {};

  const float* dptr = data + (size_t)b * NC * HW;

  for (int k0 = 0; k0 < HW; k0 += KT) {
    __syncthreads();
    // cooperative coalesced load: 256 rows x 32 floats = 2048 float4
    #pragma unroll
    for (int it = 0; it < 8; ++it) {
      int idx = tid + it * 256;
      int row = idx >> 3;      // 8 float4 per row
      int c4  = idx & 7;
      float4 v = *(const float4*)(dptr + (size_t)row * HW + k0 + c4 * 4);
      *(float4*)(&slab[row * SLAB_STRIDE + c4 * 4]) = v;
    }
    __syncthreads();

    for (int kk = 0; kk < KT; kk += 4) {
      v2f a;
      a.x = slab[rowA * SLAB_STRIDE + kk + koff];
      a.y = slab[rowA * SLAB_STRIDE + kk + koff + 1];
      #pragma unroll
      for (int j = 0; j < 4; ++j) {
        int rowB = (jBase + j) * 16 + laneM;
        v2f bf;
        bf.x = slab[rowB * SLAB_STRIDE + kk + koff];
        bf.y = slab[rowB * SLAB_STRIDE + kk + koff + 1];
        acc[j] = __builtin_amdgcn_wmma_f32_16x16x4_f32(
            false, a, false, bf, (short)0, acc[j], false, false);
      }
    }
  }

  // D layout: reg r -> M = r + 8*(lane>=16), N = lane&15
  const int Mbase = bi * 32 + iTile * 16 + 8 * (lane >> 4);
  const int Nidx  = lane & 15;
  for (int j = 0; j < 4; ++j) {
    int col = (jBase + j) * 16 + Nidx;
    #pragma unroll
    for (int r = 0; r < 8; ++r)
      atomicAdd(&G[(size_t)(Mbase + r) * NC + col], acc[j][r]);
  }
}

// ---------------------------------------------------------------------------
// 3. cov = (G - N mu mu^T) / (N-1);  V = I
// ---------------------------------------------------------------------------
__global__ __launch_bounds__(256) void finalize_kernel(const float* __restrict__ G,
                                                       const float* __restrict__ mu,
                                                       float* __restrict__ A,
                                                       float* __restrict__ V) {
  const int i = blockIdx.x;
  const int j = threadIdx.x;
  const size_t idx = (size_t)i * NC + j;
  A[idx] = (G[idx] - (float)NTOT * mu[i] * mu[j]) / (float)(NTOT - 1);
  V[idx] = (i == j) ? 1.f : 0.f;
}

// ---------------------------------------------------------------------------
// 4. Parallel cyclic Jacobi eigensolver, single workgroup (1024 threads).
//    A, V live in global memory (256KB each -> L2 resident).
// ---------------------------------------------------------------------------
__global__ __launch_bounds__(1024) void jacobi_kernel(float* __restrict__ A,
                                                      float* __restrict__ V) {
  __shared__ int   top[128], bot[128], ntop[128], nbot[128];
  __shared__ float cs[128], sn[128];
  const int tid = threadIdx.x;

  if (tid < 128) { top[tid] = 2 * tid; bot[tid] = 2 * tid + 1; }
  __syncthreads();

  for (int sweep = 0; sweep < 12; ++sweep) {
    for (int round = 0; round < 255; ++round) {
      // rotation parameters for 128 disjoint pairs
      if (tid < 128) {
        int p = top[tid], q = bot[tid];
        if (p > q) { int t = p; p = q; q = t; }
        float app = A[(size_t)p * NC + p];
        float aqq = A[(size_t)q * NC + q];
        float apq = A[(size_t)p * NC + q];
        float c = 1.f, s = 0.f;
        if (fabsf(apq) > 1e-20f) {
          float tau = (aqq - app) / (2.f * apq);
          float t = (tau >= 0.f ? 1.f : -1.f) /
                    (fabsf(tau) + sqrtf(1.f + tau * tau));
          c = rsqrtf(1.f + t * t);
          s = t * c;
        }
        cs[tid] = c; sn[tid] = s;
      }
      __threadfence();
      __syncthreads();

      // row update: B = J^T A
      for (int idx = tid; idx < 128 * 256; idx += 1024) {
        int pr = idx >> 8, j = idx & 255;
        int p = top[pr], q = bot[pr];
        if (p > q) { int t = p; p = q; q = t; }
        float c = cs[pr], s = sn[pr];
        float ap = A[(size_t)p * NC + j], aq = A[(size_t)q * NC + j];
        A[(size_t)p * NC + j] = c * ap - s * aq;
        A[(size_t)q * NC + j] = s * ap + c * aq;
      }
      __threadfence();
      __syncthreads();

      // column update: A = B J ; V = V J
      for (int idx = tid; idx < 128 * 256; idx += 1024) {
        int pr = idx >> 8, j = idx & 255;
        int p = top[pr], q = bot[pr];
        if (p > q) { int t = p; p = q; q = t; }
        float c = cs[pr], s = sn[pr];
        float ap = A[(size_t)j * NC + p], aq = A[(size_t)j * NC + q];
        A[(size_t)j * NC + p] = c * ap - s * aq;
        A[(size_t)j * NC + q] = s * ap + c * aq;
        float vp = V[(size_t)j * NC + p], vq = V[(size_t)j * NC + q];
        V[(size_t)j * NC + p] = c * vp - s * vq;
        V[(size_t)j * NC + q] = s * vp + c * vq;
      }
      __threadfence();
      __syncthreads();

      // tournament rotation (fixes slot top[0])
      if (tid < 128) {
        int nt, nb;
        if (tid == 0)      nt = top[0];
        else if (tid == 1) nt = bot[0];
        else               nt = top[tid - 1];
        if (tid < 127)     nb = bot[tid + 1];
        else               nb = top[127];
        ntop[tid] = nt; nbot[tid] = nb;
      }
      __syncthreads();
      if (tid < 128) { top[tid] = ntop[tid]; bot[tid] = nbot[tid]; }
      __syncthreads();
    }
  }
}

// ---------------------------------------------------------------------------
// 5. Rank eigenvalues (descending) and gather top-128 eigenvectors:
//    P[c*128 + rank] = V[c*256 + i]
// ---------------------------------------------------------------------------
__global__ __launch_bounds__(256) void pick_kernel(const float* __restrict__ A,
                                                   const float* __restrict__ V,
                                                   float* __restrict__ P) {
  __shared__ float lam[256];
  const int tid = threadIdx.x;
  lam[tid] = A[(size_t)tid * NC + tid];
  __syncthreads();
  const float li = lam[tid];
  int rank = 0;
  for (int j = 0; j < 256; ++j) {
    float lj = lam[j];
    rank += (lj > li) || (lj == li && j < tid);
  }
  if (rank < KOUT) {
    for (int c = 0; c < NC; ++c)
      P[(size_t)c * KOUT + rank] = V[(size_t)c * NC + tid];
  }
}

// ---------------------------------------------------------------------------
// 6. Projection: out_b (128 x 4096) = P^T (128x256) * (X_b - mu) via f32 WMMA.
//    Block: 8 waves; wave w owns output tile-row w, 4 tile-cols (64 hw cols).
// ---------------------------------------------------------------------------
__global__ __launch_bounds__(256) void proj_kernel(const float* __restrict__ data,
                                                   const float* __restrict__ mu,
                                                   const float* __restrict__ P,
                                                   float* __restrict__ out) {
  const int hw0 = blockIdx.x * 64;
  const int b   = blockIdx.y;
  const int tid = threadIdx.x;
  const int wave = tid >> 5;
  const int lane = tid & 31;
  const int laneM = lane & 15;
  const int koff  = (lane >> 4) * 2;

  v8f acc[4] = {};
  const float* dptr = data + (size_t)b * NC * HW;
  const int mrow = wave * 16 + laneM;   // output row for A fragment

  for (int k0 = 0; k0 < NC; k0 += 4) {
    const int c0 = k0 + koff;
    v2f a;
    a.x = P[(size_t)c0 * KOUT + mrow];
    a.y = P[(size_t)(c0 + 1) * KOUT + mrow];
    const float mu0 = mu[c0];
    const float mu1 = mu[c0 + 1];
    #pragma unroll
    for (int j = 0; j < 4; ++j) {
      int col = hw0 + j * 16 + laneM;
      v2f bf;
      bf.x = dptr[(size_t)c0 * HW + col] - mu0;
      bf.y = dptr[(size_t)(c0 + 1) * HW + col] - mu1;
      acc[j] = __builtin_amdgcn_wmma_f32_16x16x4_f32(
          false, a, false, bf, (short)0, acc[j], false, false);
    }
  }

  const int Mbase = wave * 16 + 8 * (lane >> 4);
  const int Nl    = lane & 15;
  float* optr = out + (size_t)b * KOUT * HW;
  for (int j = 0; j < 4; ++j) {
    int col = hw0 + j * 16 + Nl;
    #pragma unroll
    for (int r = 0; r < 8; ++r)
      optr[(size_t)(Mbase + r) * HW + col] = acc[j][r];
  }
}

// ---------------------------------------------------------------------------
extern "C" void kernel_launch(void* const* d_in, const int* in_sizes, int n_in,
                              void* d_out, int out_size, void* d_ws, size_t ws_size,
                              hipStream_t stream) {
  const float* data = (const float*)d_in[0];
  float* out = (float*)d_out;
  float* ws  = (float*)d_ws;

  // workspace layout (floats)
  float* mu   = ws;                 // 256
  float* G    = mu + 256;           // 65536
  float* Amat = G + 65536;          // 65536
  float* Vmat = Amat + 65536;       // 65536
  float* P    = Vmat + 65536;       // 256*128 = 32768
  // total: 229,632 floats = ~0.92 MB

  mean_kernel    <<<256, 256, 0, stream>>>(data, mu);
  zero_kernel    <<<256, 256, 0, stream>>>(G);
  gram_kernel    <<<dim3(8, 32), 256, 0, stream>>>(data, G);
  finalize_kernel<<<256, 256, 0, stream>>>(G, mu, Amat, Vmat);
  jacobi_kernel  <<<1, 1024, 0, stream>>>(Amat, Vmat);
  pick_kernel    <<<1, 256, 0, stream>>>(Amat, Vmat, P);
  proj_kernel    <<<dim3(64, 32), 256, 0, stream>>>(data, mu, P, out);
}